// AFNO2D_48661979464408
// MI455X (gfx1250) — compile-verified
//
#include <hip/hip_runtime.h>

// ---------------------------------------------------------------------------
// AFNO2D on MI455X (gfx1250, wave32)
//   DHT2D via separable 128x128 basis GEMMs (cas identity), WMMA f32 16x16x4.
//   B-operand strips staged to LDS by the Tensor Data Mover (TDM), D# built
//   per CDNA5 ISA ch.8; waits via s_wait_tensorcnt. An asm memory-clobber
//   anchors the TDM's LDS writes for the compiler's alias analysis.
//   Block-MLP: 8 waves = 8 channel blocks, o1 re-fragmented through LDS.
// ---------------------------------------------------------------------------

typedef __attribute__((ext_vector_type(2))) float v2f;
typedef __attribute__((ext_vector_type(8))) float v8f;
typedef __attribute__((ext_vector_type(4))) unsigned int u32x4;
typedef __attribute__((ext_vector_type(8))) int i32x8;
typedef __attribute__((ext_vector_type(4))) int i32x4;

#define B_   2
#define H_   128
#define W_   128
#define N_   (H_ * W_)          // 16384
#define C_   768
#define NB_  8
#define BS_  96                 // channels per block

static __device__ __forceinline__ v8f wmma_f32(v2f a, v2f b, v8f c) {
  // 8 args: (neg_a, A, neg_b, B, c_mod, C, reuse_a, reuse_b)
  return __builtin_amdgcn_wmma_f32_16x16x4_f32(
      false, a, false, b, (short)0, c, false, false);
}

// A fragment (16x4 f32): lanes 0-15 -> row M=lane, K={kk,kk+1};
// lanes 16-31 -> row M=lane-16, K={kk+2,kk+3}. Row-major, stride sa.
static __device__ __forceinline__ v2f
load_afrag(const float* __restrict__ A, int sa, int row0, int kk, int lr, int hi) {
  const float* p = A + (size_t)(row0 + lr) * sa + (kk + hi * 2);
  v2f a; a.x = p[0]; a.y = p[1];
  return a;
}

// B fragment (4x16 f32): VGPR0 -> rows K={kk, kk+2}, VGPR1 -> K+1; N = lane%16.
static __device__ __forceinline__ v2f
load_bfrag(const float* __restrict__ Bm, int sb, int kk, int col0, int lr, int hi) {
  const float* p = Bm + (size_t)(kk + hi * 2) * sb + (col0 + lr);
  v2f b; b.x = p[0]; b.y = p[sb];
  return b;
}

// Low 32 bits of a generic pointer into LDS == LDS byte offset (ISA 10.2).
static __device__ __forceinline__ unsigned lds_offset_of(const void* p) {
  return (unsigned)(unsigned long long)p;
}

// Compiler fence: pretend LDS behind `p` was written by an opaque agent (the
// TDM). Keeps the array allocated and orders subsequent ds_loads after it.
static __device__ __forceinline__ void lds_publish(void* p) {
  asm volatile("" : : "v"(p) : "memory");
}

// ---------------------------------------------------------------------------
// TDM: async-DMA a 2D f32 tile (tile_w contiguous elems x tile_h rows,
// row stride row_stride elems) from global into LDS at byte offset lds_off.
// D# layout per CDNA5 ISA ch.8 (group0: count/lds/global/type=2;
// group1: data_size=4B, tensor dims, tile dims, dim0 stride).
// ---------------------------------------------------------------------------
static __device__ __forceinline__ void tdm_load_2d_f32(
    const float* gptr, unsigned lds_off, unsigned tile_w, unsigned tile_h,
    unsigned row_stride, unsigned tensor_w, unsigned tensor_h) {
  unsigned long long ga = (unsigned long long)(const void*)gptr;
  u32x4 g0;
  g0[0] = 1u;                                   // count=1 (valid), user mode
  g0[1] = lds_off;                              // lds_addr (bytes)
  g0[2] = (unsigned)ga;                         // global_addr[31:0]
  g0[3] = ((unsigned)(ga >> 32) & 0x01FFFFFFu)  // global_addr[56:32]
          | (2u << 30);                         // type = 2 ("image")
  i32x8 g1;
  g1[0] = (int)(2u << 16);                      // data_size = 2 -> 4 bytes
  g1[1] = (int)((tensor_w & 0xFFFFu) << 16);    // tensor_dim0[15:0]
  g1[2] = (int)(((tensor_w >> 16) & 0xFFFFu) |  // tensor_dim0[31:16]
                ((tensor_h & 0xFFFFu) << 16));  // tensor_dim1[15:0]
  g1[3] = (int)(((tensor_h >> 16) & 0xFFFFu) |  // tensor_dim1[31:16]
                ((tile_w & 0xFFFFu) << 16));    // tile_dim0
  g1[4] = (int)(tile_h & 0xFFFFu);              // tile_dim1 (tile_dim2 = 0)
  g1[5] = (int)row_stride;                      // tensor_dim0_stride[31:0]
  g1[6] = 0;                                    // stride0[47:32], stride1[15:0]
  g1[7] = 0;                                    // stride1[47:16]
  i32x4 z4 = {0, 0, 0, 0};
  i32x8 z8 = {0, 0, 0, 0, 0, 0, 0, 0};
  __builtin_amdgcn_tensor_load_to_lds(g0, g1, z4, z4, z8, 0);
}

// ---------------------------------------------------------------------------
// Basis: cos, sin, cas+ = cos+sin, cas- = cos-sin, each 128x128.
// ---------------------------------------------------------------------------
__global__ __launch_bounds__(256) void afno_init_basis(float* __restrict__ basis) {
  int t = blockIdx.x * blockDim.x + threadIdx.x;
  if (t >= H_ * H_) return;
  int i = t / H_, j = t % H_;
  float ang = (6.283185307179586f / (float)H_) * (float)((i * j) % H_);
  float s, c;
  __sincosf(ang, &s, &c);
  basis[t]               = c;      // cos
  basis[H_ * H_ + t]     = s;      // sin
  basis[2 * H_ * H_ + t] = c + s;  // cas+
  basis[3 * H_ * H_ + t] = c - s;  // cas-
}

// ---------------------------------------------------------------------------
// Pass A (along W): WG = (slab b,m) x (c-tile). TDM stages X[:, c0:c0+16]
// ([128 x 16] = 8KB) into LDS once; 8 waves = 8 l-tiles share it.
//   U_t[b,l,m,:] = cas+ @ X,  V_t[b,l,m,:] = cas- @ X  (spatially transposed)
// ---------------------------------------------------------------------------
__global__ __launch_bounds__(256) void afno_dht_pass_a(
    const float* __restrict__ src, float* __restrict__ dstU,
    float* __restrict__ dstV, const float* __restrict__ casP,
    const float* __restrict__ casM) {
  __shared__ float xs[W_ * 16];                 // [K row][16 cols]
  const int lane = threadIdx.x & 31;
  const int wave = threadIdx.x >> 5;
  const int lr = lane & 15, hi = lane >> 4;

  const int slab  = blockIdx.x / 48;            // b*128 + m
  const int ctile = blockIdx.x % 48;
  const int b = slab >> 7, m = slab & 127;
  const int col0 = ctile * 16;

  const float* X = src + ((size_t)b * N_ + (size_t)m * W_) * C_;

  if (wave == 0) {
    tdm_load_2d_f32(X + col0, lds_offset_of(xs), 16u, (unsigned)W_,
                    (unsigned)C_, (unsigned)(C_ - col0), (unsigned)W_);
    __builtin_amdgcn_s_wait_tensorcnt(0);
  }
  __syncthreads();
  lds_publish(xs);

  const int row0 = wave * 16;                   // l-tile of this wave
  v8f accU = {}, accV = {};
#pragma unroll 4
  for (int kk = 0; kk < W_; kk += 4) {
    v2f aP = load_afrag(casP, H_, row0, kk, lr, hi);
    v2f aM = load_afrag(casM, H_, row0, kk, lr, hi);
    const int k0 = kk + hi * 2;
    v2f bx; bx.x = xs[k0 * 16 + lr]; bx.y = xs[(k0 + 1) * 16 + lr];
    accU = wmma_f32(aP, bx, accU);
    accV = wmma_f32(aM, bx, accV);
  }

#pragma unroll
  for (int v = 0; v < 8; ++v) {
    int l = row0 + v + hi * 8;
    size_t o = (((size_t)b * H_ + l) * W_ + m) * C_ + col0 + lr;
    dstU[o] = accU[v];
    dstV[o] = accV[v];
  }
}

// ---------------------------------------------------------------------------
// Pass B (along H): WG = (slab b,l) x (c-tile). Two TDMs stage the U and V
// strips (2 x 8KB); dst[b,k,l,:] = scale*(cos@U + sin@V) (+ bias[b,k*W+l,:]).
// ---------------------------------------------------------------------------
__global__ __launch_bounds__(256) void afno_dht_pass_b(
    const float* __restrict__ srcU, const float* __restrict__ srcV,
    float* __restrict__ dst, const float* __restrict__ cosB,
    const float* __restrict__ sinB, float scale,
    const float* __restrict__ bias) {
  __shared__ float uv[2 * W_ * 16];             // U strip, then V strip
  const int lane = threadIdx.x & 31;
  const int wave = threadIdx.x >> 5;
  const int lr = lane & 15, hi = lane >> 4;

  const int slab  = blockIdx.x / 48;            // b*128 + l
  const int ctile = blockIdx.x % 48;
  const int b = slab >> 7, l = slab & 127;
  const int col0 = ctile * 16;

  const float* U = srcU + ((size_t)b * H_ + l) * W_ * C_;
  const float* V = srcV + ((size_t)b * H_ + l) * W_ * C_;

  if (wave == 0) {
    tdm_load_2d_f32(U + col0, lds_offset_of(uv), 16u, (unsigned)H_,
                    (unsigned)C_, (unsigned)(C_ - col0), (unsigned)H_);
    tdm_load_2d_f32(V + col0,
                    lds_offset_of(uv) + (unsigned)(W_ * 16 * sizeof(float)),
                    16u, (unsigned)H_, (unsigned)C_, (unsigned)(C_ - col0),
                    (unsigned)H_);
    __builtin_amdgcn_s_wait_tensorcnt(0);
  }
  __syncthreads();
  lds_publish(uv);

  const float* us = uv;
  const float* vs = uv + W_ * 16;
  const int row0 = wave * 16;                   // k-tile of this wave

  v8f acc = {};
#pragma unroll 4
  for (int kk = 0; kk < H_; kk += 4) {
    v2f aC = load_afrag(cosB, H_, row0, kk, lr, hi);
    v2f aS = load_afrag(sinB, H_, row0, kk, lr, hi);
    const int k0 = kk + hi * 2;
    v2f bU; bU.x = us[k0 * 16 + lr]; bU.y = us[(k0 + 1) * 16 + lr];
    v2f bV; bV.x = vs[k0 * 16 + lr]; bV.y = vs[(k0 + 1) * 16 + lr];
    acc = wmma_f32(aC, bU, acc);
    acc = wmma_f32(aS, bV, acc);
  }

#pragma unroll
  for (int v = 0; v < 8; ++v) {
    int k = row0 + v + hi * 8;
    size_t n = (size_t)k * W_ + l;
    size_t o = ((size_t)b * N_ + n) * C_ + col0 + lr;
    float r = acc[v] * scale;
    if (bias) r += bias[o];
    dst[o] = r;
  }
}

// ---------------------------------------------------------------------------
// Block MLP over frequency points. One WG = 16 spatial points; wave j = block j.
//   o1 = relu(Xk_b @ W1[0]) + relu(Xnk_b @ W1[1])
//   o2 = o1 @ W2[0] + relu(Xnk_b @ W1[1]) @ W2[1]
// Xnk rows gathered via periodic flip; o1/o1_nk re-fragmented through LDS.
// ---------------------------------------------------------------------------
__global__ __launch_bounds__(256) void afno_mlp(
    const float* __restrict__ Xk, const float* __restrict__ w1,
    const float* __restrict__ w2, float* __restrict__ o2) {
  extern __shared__ float smem[];  // 8 waves * 2 * 16*96 floats = 96 KiB
  const int lane = threadIdx.x & 31;
  const int wave = threadIdx.x >> 5;  // channel block j
  const int lr = lane & 15, hi = lane >> 4;

  const int mt = blockIdx.x;          // 0..2047
  const int b  = mt >> 10;            // 1024 M-tiles per batch image
  const int p0 = (mt & 1023) * 16;
  const int j  = wave;

  const float* W10 = w1 + (size_t)j * BS_ * BS_;
  const float* W11 = w1 + (size_t)(NB_ + j) * BS_ * BS_;
  const float* W20 = w2 + (size_t)j * BS_ * BS_;
  const float* W21 = w2 + (size_t)(NB_ + j) * BS_ * BS_;

  float* o1s  = smem + (size_t)wave * 2 * 16 * BS_;
  float* o1ns = o1s + 16 * BS_;

  const int p  = p0 + lr;             // this lane's spatial row for A frags
  const int fp = (N_ - p) & (N_ - 1); // periodic flip (-p mod 16384)
  const float* rowK  = Xk + ((size_t)b * N_ + p)  * C_ + j * BS_;
  const float* rowNK = Xk + ((size_t)b * N_ + fp) * C_ + j * BS_;

  // ---- layer 1: two 16x96 @ 96x96 GEMMs, relu, stash in LDS ----
  for (int nt = 0; nt < 6; ++nt) {
    v8f a1k = {}, a1n = {};
#pragma unroll 4
    for (int kk = 0; kk < BS_; kk += 4) {
      const int k0 = kk + hi * 2;
      v2f ak; ak.x = rowK[k0];  ak.y = rowK[k0 + 1];
      v2f an; an.x = rowNK[k0]; an.y = rowNK[k0 + 1];
      v2f b0 = load_bfrag(W10, BS_, kk, nt * 16, lr, hi);
      v2f b1 = load_bfrag(W11, BS_, kk, nt * 16, lr, hi);
      a1k = wmma_f32(ak, b0, a1k);
      a1n = wmma_f32(an, b1, a1n);
    }
#pragma unroll
    for (int v = 0; v < 8; ++v) {
      float rk = fmaxf(a1k[v], 0.0f);
      float rn = fmaxf(a1n[v], 0.0f);
      int row = v + hi * 8, col = nt * 16 + lr;
      o1s[row * BS_ + col]  = rk + rn;  // o1
      o1ns[row * BS_ + col] = rn;       // relu(o1_nk)
    }
  }
  __syncthreads();

  // ---- layer 2: o2 = o1 @ W2[0] + o1_nk @ W2[1] ----
  for (int nt = 0; nt < 6; ++nt) {
    v8f acc = {};
#pragma unroll 4
    for (int kk = 0; kk < BS_; kk += 4) {
      v2f aO = load_afrag(o1s, BS_, 0, kk, lr, hi);
      v2f aN = load_afrag(o1ns, BS_, 0, kk, lr, hi);
      v2f b0 = load_bfrag(W20, BS_, kk, nt * 16, lr, hi);
      v2f b1 = load_bfrag(W21, BS_, kk, nt * 16, lr, hi);
      acc = wmma_f32(aO, b0, acc);
      acc = wmma_f32(aN, b1, acc);
    }
#pragma unroll
    for (int v = 0; v < 8; ++v) {
      int row = p0 + v + hi * 8, col = nt * 16 + lr;
      o2[((size_t)b * N_ + row) * C_ + j * BS_ + col] = acc[v];
    }
  }
}

// ---------------------------------------------------------------------------
// Launch: basis -> fwd passA -> fwd passB -> MLP -> inv passA -> inv passB(+bias)
// Workspace: [basis 256KB][buf0 96MiB][buf1 96MiB][buf2 96MiB]
// ---------------------------------------------------------------------------
extern "C" void kernel_launch(void* const* d_in, const int* in_sizes, int n_in,
                              void* d_out, int out_size, void* d_ws,
                              size_t ws_size, hipStream_t stream) {
  const float* x  = (const float*)d_in[0];
  const float* w1 = (const float*)d_in[1];
  const float* w2 = (const float*)d_in[2];
  float* y = (float*)d_out;

  char* ws = (char*)d_ws;
  float* basis = (float*)ws;
  const size_t MAT = (size_t)H_ * H_;
  float* cosB = basis;
  float* sinB = basis + MAT;
  float* casP = basis + 2 * MAT;
  float* casM = basis + 3 * MAT;

  const size_t BUF = (size_t)B_ * N_ * C_;       // 25.2M elements
  float* buf0 = (float*)(ws + (1 << 20));
  float* buf1 = buf0 + BUF;
  float* buf2 = buf1 + BUF;

  afno_init_basis<<<(H_ * H_ + 255) / 256, 256, 0, stream>>>(basis);

  const int gDHT = (B_ * H_) * 48;               // 12288 WGs of 8 waves
  const size_t mlpLds = (size_t)NB_ * 2 * 16 * BS_ * sizeof(float);  // 96 KiB

  // forward DHT:  x -> U(buf0), V(buf1) -> Xk(buf2)
  afno_dht_pass_a<<<gDHT, 256, 0, stream>>>(x, buf0, buf1, casP, casM);
  afno_dht_pass_b<<<gDHT, 256, 0, stream>>>(buf0, buf1, buf2, cosB, sinB, 1.0f,
                                            nullptr);
  // block MLP:    Xk(buf2) -> o2(buf0)
  afno_mlp<<<B_ * (N_ / 16), 256, mlpLds, stream>>>(buf2, w1, w2, buf0);
  // inverse DHT:  o2(buf0) -> U(buf1), V(buf2) -> y (+ residual x), / (H*W)
  afno_dht_pass_a<<<gDHT, 256, 0, stream>>>(buf0, buf1, buf2, casP, casM);
  afno_dht_pass_b<<<gDHT, 256, 0, stream>>>(buf1, buf2, y, cosB, sinB,
                                            1.0f / (float)N_, x);
}